// _ConvBlock7_43018392436870
// MI455X (gfx1250) — compile-verified
//
#include <hip/hip_runtime.h>

// ---------------------------------------------------------------------------
// MI455X (gfx1250) implementation of:
//   h = x@W_root + b ; m = x@W_nbr
//   h += segment_sum(m[src] * edge_attr, dst)           (E = 3.2M edges)
//   y  = segment_sum(h[pool_src] * pool_attr, pool_dst) (EP = 2.4M edges)
//
// Phase 1 is dense fp32 GEMM -> V_WMMA_F32_16X16X4_F32 (exact fp32 matrix op).
// Phases 2/3 are atomic-scatter bound (~2 GB traffic, ~350M f32 atomics);
// we use float4 gathers (GLOBAL_LOAD_B128) + unsafeAtomicAdd
// (GLOBAL_ATOMIC_ADD_F32, no CAS loop).
//
// Workspace layout (d_ws): h [Nc*64 f32] | m [Nc*64 f32]  (~102.4 MB total)
// ---------------------------------------------------------------------------

#define C_IN  128
#define C_OUT 64

typedef __attribute__((ext_vector_type(2))) float v2f;
typedef __attribute__((ext_vector_type(8))) float v8f;

// ---- Phase 1: dual GEMM via fp32 WMMA ------------------------------------
// Block = 256 threads = 8 waves. Wave pair (2w) handles one 16-row tile:
//   even wave -> h = x@W_root + b, odd wave -> m = x@W_nbr.
// Each wave produces the full 16x64 tile (4 x v8f accumulators), K-loop of 32
// steps of V_WMMA_F32_16X16X4_F32 (128 wmma per wave).
__global__ __launch_bounds__(256)
void gemm_dual_wmma(const float* __restrict__ x,
                    const float* __restrict__ Wroot,
                    const float* __restrict__ Wnbr,
                    const float* __restrict__ bias,
                    float* __restrict__ h,
                    float* __restrict__ m,
                    int nc)
{
    const int lane    = threadIdx.x & 31;
    const int wave    = threadIdx.x >> 5;
    const int half    = lane >> 4;      // 0: lanes 0-15, 1: lanes 16-31
    const int r       = lane & 15;
    const int rowBase = (blockIdx.x * 4 + (wave >> 1)) * 16;
    const bool isH    = (wave & 1) == 0;

    const float* __restrict__ W   = isH ? Wroot : Wnbr;
    float* __restrict__       out = isH ? h : m;

    // Clamp A-row for tail safety (arithmetic, no divergence: WMMA needs EXEC=all-1s)
    int arow = rowBase + r;
    if (arow >= nc) arow = nc - 1;
    const float* __restrict__ xrow = x + (size_t)arow * C_IN;

    v8f acc0 = {}, acc1 = {}, acc2 = {}, acc3 = {};

    // A 16x4 fp32 layout (ISA 7.12.2): lanes 0-15 hold K={0,1}, lanes 16-31 K={2,3}
    // B 4x16 fp32 mirrored: lane col = r, VGPR0 = row k+2*half, VGPR1 = row k+2*half+1
#pragma unroll
    for (int k = 0; k < C_IN; k += 4) {
        v2f a;
        a.x = xrow[k + 2 * half + 0];
        a.y = xrow[k + 2 * half + 1];

        const float* __restrict__ wp = W + (size_t)(k + 2 * half) * C_OUT + r;
        v2f b0, b1, b2, b3;
        b0.x = wp[0];          b0.y = wp[C_OUT];
        b1.x = wp[16];         b1.y = wp[C_OUT + 16];
        b2.x = wp[32];         b2.y = wp[C_OUT + 32];
        b3.x = wp[48];         b3.y = wp[C_OUT + 48];

        acc0 = __builtin_amdgcn_wmma_f32_16x16x4_f32(false, a, false, b0, (short)0, acc0, false, false);
        acc1 = __builtin_amdgcn_wmma_f32_16x16x4_f32(false, a, false, b1, (short)0, acc1, false, false);
        acc2 = __builtin_amdgcn_wmma_f32_16x16x4_f32(false, a, false, b2, (short)0, acc2, false, false);
        acc3 = __builtin_amdgcn_wmma_f32_16x16x4_f32(false, a, false, b3, (short)0, acc3, false, false);
    }

    // C/D layout: VGPR v, lanes 0-15 -> (row v,   col r)
    //                      lanes 16-31 -> (row v+8, col r)
    v8f accs[4] = {acc0, acc1, acc2, acc3};
#pragma unroll
    for (int t = 0; t < 4; ++t) {
        const int col = 16 * t + r;
        const float bv = isH ? bias[col] : 0.0f;
#pragma unroll
        for (int v = 0; v < 8; ++v) {
            const int row = rowBase + v + 8 * half;
            if (row < nc)
                out[(size_t)row * C_OUT + col] = accs[t][v] + bv;
        }
    }
}

// ---- Phase 2: edge message scatter: h[dst] += m[src] * w ------------------
// 16 threads per edge, 4 channels per thread (float4 gather + 4 f32 atomics).
__global__ __launch_bounds__(256)
void edge_scatter(const int* __restrict__ edge_index, // [2, E]
                  const float* __restrict__ eattr,    // [E]
                  const float* __restrict__ m,        // [Nc, 64]
                  float* __restrict__ h,              // [Nc, 64]
                  int E)
{
    const long long idx = (long long)blockIdx.x * blockDim.x + threadIdx.x;
    const long long e   = idx >> 4;
    if (e >= E) return;
    const int cg = (int)(idx & 15) * 4;

    const int   s = edge_index[e];
    const int   d = edge_index[(long long)E + e];
    const float w = eattr[e];

    const float4 mv = *(const float4*)(m + (size_t)s * C_OUT + cg);
    float* __restrict__ hp = h + (size_t)d * C_OUT + cg;
    unsafeAtomicAdd(hp + 0, mv.x * w);
    unsafeAtomicAdd(hp + 1, mv.y * w);
    unsafeAtomicAdd(hp + 2, mv.z * w);
    unsafeAtomicAdd(hp + 3, mv.w * w);
}

// ---- Phase 3: unpool scatter: y[pool_dst] += h[pool_src] * w --------------
__global__ __launch_bounds__(256)
void unpool_scatter(const int* __restrict__ psrc,
                    const int* __restrict__ pdst,
                    const float* __restrict__ pattr,
                    const float* __restrict__ h,
                    float* __restrict__ y,
                    int EP)
{
    const long long idx = (long long)blockIdx.x * blockDim.x + threadIdx.x;
    const long long e   = idx >> 4;
    if (e >= EP) return;
    const int cg = (int)(idx & 15) * 4;

    const int   s = psrc[e];
    const int   d = pdst[e];
    const float w = pattr[e];

    const float4 hv = *(const float4*)(h + (size_t)s * C_OUT + cg);
    float* __restrict__ yp = y + (size_t)d * C_OUT + cg;
    unsafeAtomicAdd(yp + 0, hv.x * w);
    unsafeAtomicAdd(yp + 1, hv.y * w);
    unsafeAtomicAdd(yp + 2, hv.z * w);
    unsafeAtomicAdd(yp + 3, hv.w * w);
}

extern "C" void kernel_launch(void* const* d_in, const int* in_sizes, int n_in,
                              void* d_out, int out_size, void* d_ws, size_t ws_size,
                              hipStream_t stream) {
    // setup_inputs() order:
    // 0:x 1:W_root 2:W_nbr 3:b 4:edge_index 5:edge_attr
    // 6:pool_src 7:pool_dst 8:pool_edge_attr 9:n_fine
    const float* x     = (const float*)d_in[0];
    const float* Wroot = (const float*)d_in[1];
    const float* Wnbr  = (const float*)d_in[2];
    const float* bias  = (const float*)d_in[3];
    const int*   eidx  = (const int*)d_in[4];
    const float* eattr = (const float*)d_in[5];
    const int*   psrc  = (const int*)d_in[6];
    const int*   pdst  = (const int*)d_in[7];
    const float* pattr = (const float*)d_in[8];

    const int nc = in_sizes[0] / C_IN;   // 200000
    const int E  = in_sizes[5];          // 3200000 (edge_attr is [E,1])
    const int EP = in_sizes[8];          // 2400000

    float* h = (float*)d_ws;                       // [nc, 64]
    float* m = h + (size_t)nc * C_OUT;             // [nc, 64]

    // y must start at zero (harness poisons d_out); async memset is capturable.
    hipMemsetAsync(d_out, 0, (size_t)out_size * sizeof(float), stream);

    // Phase 1: 4 row-tiles of 16 per block (nc = 200000 = 3125 * 64)
    const int gemmBlocks = (nc + 63) / 64;
    gemm_dual_wmma<<<gemmBlocks, 256, 0, stream>>>(x, Wroot, Wnbr, bias, h, m, nc);

    // Phase 2: 16 threads per edge
    const long long t2 = (long long)E * 16;
    edge_scatter<<<(unsigned)((t2 + 255) / 256), 256, 0, stream>>>(eidx, eattr, m, h, E);

    // Phase 3: 16 threads per pool edge
    const long long t3 = (long long)EP * 16;
    unpool_scatter<<<(unsigned)((t3 + 255) / 256), 256, 0, stream>>>(psrc, pdst, pattr, h, (float*)d_out, EP);
}